// FeaturePyramid_30236569764329
// MI455X (gfx1250) — compile-verified
//
#include <hip/hip_runtime.h>

#define TPB 256
#define CHUNK 32
#define SS_EPS 1e-7f

// ---- CDNA5 async global->LDS path (guarded probe; fallback keeps compile clean) ----
#if defined(__has_builtin)
#if __has_builtin(__builtin_amdgcn_global_load_async_to_lds_b32) && \
    __has_builtin(__builtin_amdgcn_s_wait_asynccnt)
#define USE_ASYNC_LDS 1
#endif
#endif
#ifndef USE_ASYNC_LDS
#define USE_ASYNC_LDS 0
#endif

typedef __attribute__((address_space(1))) int glob_i32;
typedef __attribute__((address_space(3))) int lds_i32;

__device__ __forceinline__ void stage_to_lds(const float* g, float* l) {
#if USE_ASYNC_LDS
  __builtin_amdgcn_global_load_async_to_lds_b32((glob_i32*)g, (lds_i32*)l, 0, 0);
#else
  *l = *g;
#endif
}

// Wait until at most PENDING async loads remain (async loads retire in order,
// so waiting <= 32 guarantees the previous 32-load chunk has fully landed).
template <int PENDING>
__device__ __forceinline__ void stage_wait() {
#if USE_ASYNC_LDS
  __builtin_amdgcn_s_wait_asynccnt(PENDING);
  asm volatile("" ::: "memory");
#endif
}

// Non-returning agent-scope f32 atomic add -> global_atomic_add_f32 (no CAS loop)
__device__ __forceinline__ void atomAddF(float* p, float v) {
  (void)__hip_atomic_fetch_add(p, v, __ATOMIC_RELAXED, __HIP_MEMORY_SCOPE_AGENT);
}

// ---------------- softmax splat (scatter-add) body ----------------
// One thread per source pixel. Channels staged in double-buffered LDS chunks via
// async global->LDS loads, pipelined against the atomic scatter of the previous
// chunk using the ASYNCcnt counter.
// acc layout: [N, C+1, H, W] (last plane = exp-weight sum). wsum (optional): [N,H,W].
template <int C, int H, int W, bool HAS_WSUM>
__device__ __forceinline__ void splat_body(const float* __restrict__ feat, float featScale,
                                           const float* __restrict__ flow, float flowScale,
                                           const float* __restrict__ metric,
                                           float* __restrict__ acc, float* __restrict__ wsum) {
  constexpr int N = 4;
  constexpr long HW = (long)H * W;
  constexpr long TOTAL = (long)N * HW;
  constexpr int NCHUNKS = (C + CHUNK - 1) / CHUNK;

  __shared__ float sfeat[2][CHUNK * TPB];
  const int tid = threadIdx.x;
  const long pix = (long)blockIdx.x * TPB + tid;
  if (pix >= TOTAL) return;
  const int n = (int)(pix / HW);           // constant divisor -> mul/shift
  const long rem = pix - (long)n * HW;
  const int y = (int)(rem / W);
  const int x = (int)(rem - (long)y * W);

  // ---- kick chunk 0 of the channel vector into LDS (overlaps math below) ----
  const float* fbase = feat + (long)n * C * HW + rem;
  {
    constexpr int CC0 = C < CHUNK ? C : CHUNK;
#pragma unroll
    for (int k = 0; k < CC0; ++k)
      stage_to_lds(fbase + (long)k * HW, &sfeat[0][k * TPB + tid]);
  }
  __builtin_prefetch(fbase, 0, 3);  // global_prefetch_b8

  // ---- flow / metric / corner weights ----
  const float fdx = flow[((long)n * 2 + 0) * HW + rem] * flowScale;
  const float fdy = flow[((long)n * 2 + 1) * HW + rem] * flowScale;
  float mm = -metric[(long)n * HW + rem];
  mm = fminf(fmaxf(mm, -20.0f), 20.0f);
  const float e = expf(mm);

  const float gx = (float)x + fdx;
  const float gy = (float)y + fdy;
  const float x0f = floorf(gx);
  const float y0f = floorf(gy);
  const float fx = gx - x0f;
  const float fy = gy - y0f;
  const int x0 = (int)x0f;
  const int y0 = (int)y0f;

  const int cx[4] = {x0, x0 + 1, x0, x0 + 1};
  const int cy[4] = {y0, y0, y0 + 1, y0 + 1};
  float cw[4];
  cw[0] = (1.0f - fx) * (1.0f - fy);
  cw[1] = fx * (1.0f - fy);
  cw[2] = (1.0f - fx) * fy;
  cw[3] = fx * fy;
  long cidx[4]; bool cval[4];
#pragma unroll
  for (int i = 0; i < 4; ++i) {
    cval[i] = (cx[i] >= 0) && (cx[i] < W) && (cy[i] >= 0) && (cy[i] < H);
    cidx[i] = (long)cy[i] * W + cx[i];
  }

  // exp-weight plane + (optional) pure-weight sum for hole mask
  float* accw = acc + ((long)n * (C + 1) + C) * HW;
#pragma unroll
  for (int i = 0; i < 4; ++i) if (cval[i]) atomAddF(&accw[cidx[i]], e * cw[i]);
  if (HAS_WSUM) {
    float* wp = wsum + (long)n * HW;
#pragma unroll
    for (int i = 0; i < 4; ++i) if (cval[i]) atomAddF(&wp[cidx[i]], cw[i]);
  }

  // ---- pipelined channel chunks: stage(i+1) -> wait(chunk i landed) -> scatter(i) ----
  float* abase = acc + (long)n * (C + 1) * HW;
#pragma unroll
  for (int ch = 0; ch < NCHUNKS; ++ch) {
    const int base = ch * CHUNK;
    const int cc = (C - base) < CHUNK ? (C - base) : CHUNK;
    if (ch + 1 < NCHUNKS) {
      const int nbase = base + CHUNK;
      const int ncc = (C - nbase) < CHUNK ? (C - nbase) : CHUNK;
#pragma unroll
      for (int k = 0; k < ncc; ++k)
        stage_to_lds(fbase + (long)(nbase + k) * HW, &sfeat[(ch + 1) & 1][k * TPB + tid]);
      stage_wait<CHUNK>();   // prev chunk complete; next chunk still in flight
    } else {
      stage_wait<0>();
    }
#pragma unroll
    for (int k = 0; k < cc; ++k) {
      const float v = sfeat[ch & 1][k * TPB + tid] * featScale * e;
      float* ap = abase + (long)(base + k) * HW;
#pragma unroll
      for (int i = 0; i < 4; ++i) if (cval[i]) atomAddF(&ap[cidx[i]], v * cw[i]);
    }
  }
}

// ---- non-template wrappers: eagerly emitted in definition order, splat first ----
__global__ void splat_lvl0(const float* __restrict__ feat, const float* __restrict__ flow,
                           const float* __restrict__ metric, float* __restrict__ acc,
                           float* __restrict__ wsum) {
  splat_body<32, 256, 448, true>(feat, 1.0f, flow, 0.5f, metric, acc, wsum);
}
__global__ void splat_lvl1(const float* __restrict__ feat, const float* __restrict__ flow,
                           const float* __restrict__ metric, float* __restrict__ acc,
                           float* __restrict__ wsum) {
  splat_body<64, 128, 224, true>(feat, 1.0f, flow, 1.0f, metric, acc, wsum);
}
__global__ void splat_lvl2(const float* __restrict__ feat, const float* __restrict__ flow,
                           const float* __restrict__ metric, float* __restrict__ acc,
                           float* __restrict__ wsum) {
  splat_body<96, 64, 112, true>(feat, 1.0f, flow, 1.0f, metric, acc, wsum);
}
__global__ void splat_flow(const float* __restrict__ F21, const float* __restrict__ metric,
                           float* __restrict__ acc) {
  splat_body<2, 256, 448, false>(F21, 0.5f, F21, 0.5f, metric, acc, nullptr);
}

// ---------------- zero workspace ----------------
__global__ void zero_kernel(float* __restrict__ p, long n) {
  long i = (long)blockIdx.x * blockDim.x + threadIdx.x;
  if (i < n) p[i] = 0.0f;
}

// -------- 2x antialiased bilinear downsample (separable [1,3,3,1]/8) --------
template <int C, int Hs, int Ws>
__global__ void down2_kernel(const float* __restrict__ src, float* __restrict__ dst,
                             float scale) {
  constexpr int N = 4;
  constexpr int Hd = Hs >> 1, Wd = Ws >> 1;
  constexpr long TOTAL = (long)N * C * Hd * Wd;
  long i = (long)blockIdx.x * blockDim.x + threadIdx.x;
  if (i >= TOTAL) return;
  int x = (int)(i % Wd); long t = i / Wd;
  int y = (int)(t % Hd); t /= Hd;
  int c = (int)(t % C);  int n = (int)(t / C);
  const float w4[4] = {0.125f, 0.375f, 0.375f, 0.125f};
  const float* s = src + (((long)n * C + c) * (long)Hs) * Ws;
  float acc = 0.0f;
#pragma unroll
  for (int j = 0; j < 4; ++j) {
    int yy = 2 * y - 1 + j; yy = min(max(yy, 0), Hs - 1);
    float rowacc = 0.0f;
#pragma unroll
    for (int k = 0; k < 4; ++k) {
      int xx = 2 * x - 1 + k; xx = min(max(xx, 0), Ws - 1);
      rowacc += w4[k] * s[(long)yy * Ws + xx];
    }
    acc += w4[j] * rowacc;
  }
  dst[i] = acc * scale;
}

// ---------------- normalize: out = sign * acc[c] / (acc[C] + eps) ----------------
template <int C, int H, int W>
__global__ void finalize_kernel(const float* __restrict__ acc, float* __restrict__ out,
                                float sign) {
  constexpr int N = 4;
  constexpr long HW = (long)H * W;
  constexpr long TOTAL = (long)N * C * HW;
  long i = (long)blockIdx.x * blockDim.x + threadIdx.x;
  if (i >= TOTAL) return;
  const long p = i % HW;
  long t = i / HW;
  const int c = (int)(t % C);
  const int n = (int)(t / C);
  const float a = acc[((long)n * (C + 1) + c) * HW + p];
  const float d = acc[((long)n * (C + 1) + C) * HW + p];
  out[i] = sign * a / (d + SS_EPS);
}

// ---------------- hole mask: (w/(w+eps) <= 0.5) ? 1 : 0 ----------------
__global__ void mask_kernel(const float* __restrict__ wsum, float* __restrict__ out, long n) {
  long i = (long)blockIdx.x * blockDim.x + threadIdx.x;
  if (i >= n) return;
  const float w = wsum[i];
  const float q = w / (w + SS_EPS);
  out[i] = (q <= 0.5f) ? 1.0f : 0.0f;
}

static inline int nblk(long n) { return (int)((n + TPB - 1) / TPB); }

extern "C" void kernel_launch(void* const* d_in, const int* in_sizes, int n_in,
                              void* d_out, int out_size, void* d_ws, size_t ws_size,
                              hipStream_t stream) {
  (void)in_sizes; (void)n_in; (void)out_size; (void)ws_size;
  const float* x1_0 = (const float*)d_in[0];
  const float* x1_1 = (const float*)d_in[1];
  const float* x1_2 = (const float*)d_in[2];
  const float* m1t  = (const float*)d_in[6];
  const float* m2t  = (const float*)d_in[7];
  const float* F12  = (const float*)d_in[8];
  const float* F21  = (const float*)d_in[9];
  float* out = (float*)d_out;
  float* ws  = (float*)d_ws;

  const int N = 4;
  const long HW0 = 256L * 448, HW1 = 128L * 224, HW2 = 64L * 112;
  const int C0 = 32, C1 = 64, C2 = 96;

  // ---- workspace layout (floats) ----
  long off = 0;
  float* acc0  = ws + off; off += (long)N * (C0 + 1) * HW0;
  float* acc1  = ws + off; off += (long)N * (C1 + 1) * HW1;
  float* acc2  = ws + off; off += (long)N * (C2 + 1) * HW2;
  float* accF  = ws + off; off += (long)N * 3 * HW0;
  float* wsum0 = ws + off; off += (long)N * HW0;
  float* wsum1 = ws + off; off += (long)N * HW1;
  float* wsum2 = ws + off; off += (long)N * HW2;
  const long zeroTotal = off;               // everything above must start at 0
  float* flow1 = ws + off; off += (long)N * 2 * HW1;
  float* flow2 = ws + off; off += (long)N * 2 * HW2;
  float* met1  = ws + off; off += (long)N * HW1;
  float* met2  = ws + off; off += (long)N * HW2;

  // ---- output layout (floats) ----
  float* o_w0  = out;
  float* o_w1  = o_w0 + (long)N * C0 * HW0;
  float* o_w2  = o_w1 + (long)N * C1 * HW1;
  float* o_m0  = o_w2 + (long)N * C2 * HW2;
  float* o_m1  = o_m0 + (long)N * HW0;
  float* o_m2  = o_m1 + (long)N * HW1;
  float* o_Ft2 = o_m2 + (long)N * HW2;

  // 1) zero accumulators
  zero_kernel<<<nblk(zeroTotal), TPB, 0, stream>>>(ws, zeroTotal);

  // 2) pyramid downsampling of metric/flow
  //    met1 = ds(m1t); flow1 = ds(F12)*0.5(src F1t)*0.5(resize scale) = ds(F12)*0.25
  down2_kernel<1, 256, 448><<<nblk((long)N * 1 * HW1), TPB, 0, stream>>>(m1t, met1, 1.0f);
  down2_kernel<2, 256, 448><<<nblk((long)N * 2 * HW1), TPB, 0, stream>>>(F12, flow1, 0.25f);
  down2_kernel<1, 128, 224><<<nblk((long)N * 1 * HW2), TPB, 0, stream>>>(met1, met2, 1.0f);
  down2_kernel<2, 128, 224><<<nblk((long)N * 2 * HW2), TPB, 0, stream>>>(flow1, flow2, 0.5f);

  // 3) scatter splats (t = 0.5 -> F1t = 0.5*F12, m1t_s = m1t; F2t = 0.5*F21, m2t_s = m2t)
  splat_lvl0<<<nblk((long)N * HW0), TPB, 0, stream>>>(x1_0, F12, m1t, acc0, wsum0);
  splat_lvl1<<<nblk((long)N * HW1), TPB, 0, stream>>>(x1_1, flow1, met1, acc1, wsum1);
  splat_lvl2<<<nblk((long)N * HW2), TPB, 0, stream>>>(x1_2, flow2, met2, acc2, wsum2);
  splat_flow<<<nblk((long)N * HW0), TPB, 0, stream>>>(F21, m2t, accF);

  // 4) normalization + masks
  finalize_kernel<32, 256, 448><<<nblk((long)N * C0 * HW0), TPB, 0, stream>>>(acc0, o_w0, 1.0f);
  finalize_kernel<64, 128, 224><<<nblk((long)N * C1 * HW1), TPB, 0, stream>>>(acc1, o_w1, 1.0f);
  finalize_kernel<96, 64, 112><<<nblk((long)N * C2 * HW2), TPB, 0, stream>>>(acc2, o_w2, 1.0f);
  finalize_kernel<2, 256, 448><<<nblk((long)N * 2 * HW0), TPB, 0, stream>>>(accF, o_Ft2, -1.0f);
  mask_kernel<<<nblk((long)N * HW0), TPB, 0, stream>>>(wsum0, o_m0, (long)N * HW0);
  mask_kernel<<<nblk((long)N * HW1), TPB, 0, stream>>>(wsum1, o_m1, (long)N * HW1);
  mask_kernel<<<nblk((long)N * HW2), TPB, 0, stream>>>(wsum2, o_m2, (long)N * HW2);
}